// TopologicalMPLayer_78967268704334
// MI455X (gfx1250) — compile-verified
//
#include <hip/hip_runtime.h>

#define H 128
#define LEAKY 0.01f

typedef __attribute__((ext_vector_type(2))) float v2f;
typedef __attribute__((ext_vector_type(8))) float v8f;

// ---------------- utility kernels ----------------

__global__ void zero_words_kernel(int* __restrict__ p, long n) {
    long i = (long)blockIdx.x * blockDim.x + threadIdx.x;
    long stride = (long)gridDim.x * blockDim.x;
    for (; i < n; i += stride) p[i] = 0;
}

__global__ void flag_init_kernel(float* flag) { flag[0] = 0.0f; }

__global__ void flag_check_kernel(const int* __restrict__ depth_arr,
                                  const int* __restrict__ depth,
                                  int E, float* flag) {
    int i = blockIdx.x * blockDim.x + threadIdx.x;
    if (i < E && depth_arr[i] == depth[0]) flag[0] = 1.0f;  // same value from all writers
}

// ---------------- edge scatter: masked gather + segment-sum ----------------
// One wave32 per edge; lane l handles H-elements [4l, 4l+3].
// Gather is a contiguous 512B read per edge (L2-resident: feat table = 51MB < 192MB L2),
// accumulation via global_atomic_add_f32 into the [N,H] workspace accumulator.
__global__ void edge_scatter_kernel(const float* __restrict__ feat_src,
                                    const int* __restrict__ src,
                                    const int* __restrict__ dst,
                                    const int* __restrict__ depth_arr,
                                    const int* __restrict__ depth,
                                    float* __restrict__ ft,
                                    int* __restrict__ upd,
                                    int E) {
    int edge = blockIdx.x * (blockDim.x >> 5) + (threadIdx.x >> 5);
    int lane = threadIdx.x & 31;
    if (edge >= E) return;
    if (depth_arr[edge] != depth[0]) return;   // masked edge: contributes nothing
    int s = src[edge];
    int d = dst[edge];
    float4 v = *(const float4*)(feat_src + (long)s * H + lane * 4);
    float* dp = ft + (long)d * H + lane * 4;
    atomicAdd(dp + 0, v.x);
    atomicAdd(dp + 1, v.y);
    atomicAdd(dp + 2, v.z);
    atomicAdd(dp + 3, v.w);
    if (lane == 0) atomicOr(upd + d, 1);
}

// ---------------- MLP: 32 rows x (256->128 leaky_relu, 128->128) via f32 WMMA ----
// Workgroup = 8 wave32 (256 thr). Block computes rows [row0, row0+32), all 128 cols.
// Each wave owns one 16-wide N-tile and TWO 16-row M-tiles (register blocking in M):
// every B fragment (weights, global/L2) is reused by two V_WMMA_F32_16X16X4_F32,
// halving weight traffic vs. the 16-row version. A fragments come from LDS.
//
// Per ISA 7.12.2 (32-bit A 16x4): lanes 0-15 -> M=lane, V0=K0,V1=K1;
//                                  lanes 16-31 -> M=lane-16, V0=K2,V1=K3.
// B (4x16) mirrors the K split across lane halves; C/D: VGPR j -> M=j (+8 for
// upper lane half), N = lane&15.
__global__ __launch_bounds__(256) void mlp_kernel(
        const float* __restrict__ feat_dst,
        const float* __restrict__ ft,
        const int*   __restrict__ upd,
        const float* __restrict__ W1,   // [2H, H] row-major
        const float* __restrict__ b1,   // [H]
        const float* __restrict__ W2,   // [H, H] row-major
        const float* __restrict__ b2,   // [H]
        float* __restrict__ out,        // [N, H]
        int N) {
    __shared__ float Xs[32][2 * H + 4];  // stride 260 -> conflict-free column reads
    __shared__ float Hs[32][H + 4];      // stride 132 -> conflict-free column reads

    const int row0 = blockIdx.x * 32;
    const int tid  = threadIdx.x;
    const int wave = tid >> 5;
    const int lane = tid & 31;
    const int m16  = lane & 15;
    const int half = lane >> 4;
    const int n    = wave * 16 + m16;    // this lane's output column

    // Stage X = [feat_dst | ft] (32 x 256) into LDS, coalesced float4.
    for (int idx = tid; idx < 32 * (2 * H / 4); idx += 256) {
        int r  = idx >> 6;           // 64 float4 per row
        int c  = (idx & 63) * 4;
        int row = row0 + r;
        if (row >= N) row = N - 1;   // clamp (N divisible by 32 in practice)
        float4 v;
        if (c < H) v = *(const float4*)(feat_dst + (long)row * H + c);
        else       v = *(const float4*)(ft       + (long)row * H + (c - H));
        *(float4*)&Xs[r][c] = v;
    }
    __syncthreads();

    // ---- Layer 1: two 16x16 C tiles, K = 256; B fragment shared across M ----
    v8f acc0 = {0.f, 0.f, 0.f, 0.f, 0.f, 0.f, 0.f, 0.f};
    v8f acc1 = {0.f, 0.f, 0.f, 0.f, 0.f, 0.f, 0.f, 0.f};
    #pragma unroll 4
    for (int k = 0; k < 2 * H; k += 4) {
        int kk = k + 2 * half;
        v2f b;
        b.x = W1[(long)kk * H + n];
        b.y = W1[(long)(kk + 1) * H + n];
        v2f a0;
        a0.x = Xs[m16][kk];
        a0.y = Xs[m16][kk + 1];
        v2f a1;
        a1.x = Xs[16 + m16][kk];
        a1.y = Xs[16 + m16][kk + 1];
        acc0 = __builtin_amdgcn_wmma_f32_16x16x4_f32(
                   false, a0, false, b, (short)0, acc0, false, false);
        acc1 = __builtin_amdgcn_wmma_f32_16x16x4_f32(
                   false, a1, false, b, (short)0, acc1, false, false);
    }

    // bias + leaky_relu -> LDS activation tile (32 x 128)
    float bias1 = b1[n];
    #pragma unroll
    for (int j = 0; j < 8; ++j) {
        float v0 = acc0[j] + bias1;
        v0 = v0 > 0.0f ? v0 : LEAKY * v0;
        Hs[j + 8 * half][n] = v0;
        float v1 = acc1[j] + bias1;
        v1 = v1 > 0.0f ? v1 : LEAKY * v1;
        Hs[16 + j + 8 * half][n] = v1;
    }
    __syncthreads();

    // ---- Layer 2: K = 128, A from LDS activation; B fragment shared across M ----
    v8f acd0 = {0.f, 0.f, 0.f, 0.f, 0.f, 0.f, 0.f, 0.f};
    v8f acd1 = {0.f, 0.f, 0.f, 0.f, 0.f, 0.f, 0.f, 0.f};
    #pragma unroll 4
    for (int k = 0; k < H; k += 4) {
        int kk = k + 2 * half;
        v2f b;
        b.x = W2[(long)kk * H + n];
        b.y = W2[(long)(kk + 1) * H + n];
        v2f a0;
        a0.x = Hs[m16][kk];
        a0.y = Hs[m16][kk + 1];
        v2f a1;
        a1.x = Hs[16 + m16][kk];
        a1.y = Hs[16 + m16][kk + 1];
        acd0 = __builtin_amdgcn_wmma_f32_16x16x4_f32(
                   false, a0, false, b, (short)0, acd0, false, false);
        acd1 = __builtin_amdgcn_wmma_f32_16x16x4_f32(
                   false, a1, false, b, (short)0, acd1, false, false);
    }

    // bias + masked writeback: out = upd ? res : feat_dst  (both M tiles)
    float bias2 = b2[n];
    #pragma unroll
    for (int j = 0; j < 8; ++j) {
        int row = row0 + j + 8 * half;
        if (row < N) {
            float res = acd0[j] + bias2;
            float old = feat_dst[(long)row * H + n];
            out[(long)row * H + n] = upd[row] ? res : old;
        }
        int row1 = row + 16;
        if (row1 < N) {
            float res = acd1[j] + bias2;
            float old = feat_dst[(long)row1 * H + n];
            out[(long)row1 * H + n] = upd[row1] ? res : old;
        }
    }
}

// ---------------- launch ----------------

extern "C" void kernel_launch(void* const* d_in, const int* in_sizes, int n_in,
                              void* d_out, int out_size, void* d_ws, size_t ws_size,
                              hipStream_t stream) {
    const float* feat_a = (const float*)d_in[0];
    const float* feat_b = (const float*)d_in[1];
    const float* feat_c = (const float*)d_in[2];
    const int* e1_src   = (const int*)d_in[3];
    const int* e1_dst   = (const int*)d_in[4];
    const int* e1_dep   = (const int*)d_in[5];
    const int* e2_src   = (const int*)d_in[6];
    const int* e2_dst   = (const int*)d_in[7];
    const int* e2_dep   = (const int*)d_in[8];
    const float* W1_e1  = (const float*)d_in[9];
    const float* b1_e1  = (const float*)d_in[10];
    const float* W2_e1  = (const float*)d_in[11];
    const float* b2_e1  = (const float*)d_in[12];
    const float* W1_e2  = (const float*)d_in[13];
    const float* b1_e2  = (const float*)d_in[14];
    const float* W2_e2  = (const float*)d_in[15];
    const float* b2_e2  = (const float*)d_in[16];
    const int* depth    = (const int*)d_in[17];

    const int N  = in_sizes[0] / H;
    const int E1 = in_sizes[3];
    const int E2 = in_sizes[6];

    float* outA = (float*)d_out;
    float* outB = outA + (long)N * H;
    float* outC = outA + 2L * N * H;
    float* flag = outA + 3L * N * H;

    float* ft  = (float*)d_ws;                 // [N, H] f32 accumulator
    int*   upd = (int*)(ft + (long)N * H);     // [N] touched bitmap (contiguous)
    const long zwords = (long)N * H + N;

    // feat_a passthrough (unchanged by reference)
    hipMemcpyAsync(outA, feat_a, sizeof(float) * (size_t)N * H,
                   hipMemcpyDeviceToDevice, stream);

    // edge_matching_depth_found flag
    flag_init_kernel<<<1, 1, 0, stream>>>(flag);
    flag_check_kernel<<<(E1 + 255) / 256, 256, 0, stream>>>(e1_dep, depth, E1, flag);
    flag_check_kernel<<<(E2 + 255) / 256, 256, 0, stream>>>(e2_dep, depth, E2, flag);

    // ---- edge type e1: a -> b ----
    zero_words_kernel<<<2048, 256, 0, stream>>>((int*)ft, zwords);
    edge_scatter_kernel<<<(E1 + 7) / 8, 256, 0, stream>>>(
        feat_a, e1_src, e1_dst, e1_dep, depth, ft, upd, E1);
    mlp_kernel<<<(N + 31) / 32, 256, 0, stream>>>(
        feat_b, ft, upd, W1_e1, b1_e1, W2_e1, b2_e1, outB, N);

    // ---- edge type e2: b -> c (gathers from the UPDATED b, i.e. outB) ----
    zero_words_kernel<<<2048, 256, 0, stream>>>((int*)ft, zwords);
    edge_scatter_kernel<<<(E2 + 7) / 8, 256, 0, stream>>>(
        outB, e2_src, e2_dst, e2_dep, depth, ft, upd, E2);
    mlp_kernel<<<(N + 31) / 32, 256, 0, stream>>>(
        feat_c, ft, upd, W1_e2, b1_e2, W2_e2, b2_e2, outC, N);
}